// MOSEG_80547816669986
// MI455X (gfx1250) — compile-verified
//
#include <hip/hip_runtime.h>
#include <hip/hip_bf16.h>

// Problem constants (fixed by setup_inputs in the reference)
#define S_T   3
#define NPTS  483840              // 6*48*28*60 points per timestep
#define CCH   64                  // channels
#define XD    200
#define YD    200
#define NVOX  (XD*YD)             // 40000 voxels (Zd == 1)
#define CV    ((size_t)CCH*NVOX)  // 2,560,000 elems per timestep plane

typedef float f32x2 __attribute__((ext_vector_type(2)));

// Native no-return f32 atomic add at device scope: RMW resolves in L2 (the
// device coherence point; our 30.7MB accumulators are L2-resident). No-return
// form is tracked with STOREcnt; s_endpgm's implicit wait-idle fences it
// before the dependent combine kernel.
__device__ __forceinline__ void atomAddF(float* p, float v) {
    asm volatile("global_atomic_add_f32 %0, %1, off scope:SCOPE_DEV"
                 :: "v"(p), "v"(v) : "memory");
}

// ---------------------------------------------------------------------------
// Kernel 1: build per-step ego transforms (R_t, tr_t) for t = 0..S-2.
// Mirrors _euler2mat: R = xmat @ ymat @ zmat with angles (z,y,x) = ego[3:6].
// ---------------------------------------------------------------------------
__global__ void MOSEG_xform(const float* __restrict__ ego, float* __restrict__ xf) {
    if (threadIdx.x != 0 || blockIdx.x != 0) return;
    for (int t = 0; t < S_T - 1; ++t) {
        const float* e = ego + t * 6;
        float az = e[3], ay = e[4], ax = e[5];
        float cz = cosf(az), sz = sinf(az);
        float cy = cosf(ay), sy = sinf(ay);
        float cx = cosf(ax), sx = sinf(ax);
        float Zm[9] = {cz, -sz, 0.f,  sz, cz, 0.f,  0.f, 0.f, 1.f};
        float Ym[9] = {cy, 0.f, sy,   0.f, 1.f, 0.f, -sy, 0.f, cy};
        float Xm[9] = {1.f, 0.f, 0.f, 0.f, cx, -sx,  0.f, sx, cx};
        float A[9], R[9];
        for (int i = 0; i < 3; ++i)
            for (int j = 0; j < 3; ++j) {
                float s = 0.f;
                for (int k = 0; k < 3; ++k) s += Xm[i*3+k] * Ym[k*3+j];
                A[i*3+j] = s;
            }
        for (int i = 0; i < 3; ++i)
            for (int j = 0; j < 3; ++j) {
                float s = 0.f;
                for (int k = 0; k < 3; ++k) s += A[i*3+k] * Zm[k*3+j];
                R[i*3+j] = s;
            }
        for (int j = 0; j < 9; ++j) xf[t*12 + j] = R[j];
        xf[t*12 +  9] = e[0];
        xf[t*12 + 10] = e[1];
        xf[t*12 + 11] = e[2];
    }
}

// ---------------------------------------------------------------------------
// Geometry -> voxel index helper (templated so the number of sequential
// ego-transform steps is compile-time: stays in registers, no spills).
// ---------------------------------------------------------------------------
template<int NST>
__device__ __forceinline__ int voxel_index(const float (&m)[NST > 0 ? NST : 1][12],
                                           float gx, float gy, float gz) {
    #pragma unroll
    for (int k = 0; k < NST; ++k) {
        float nx = fmaf(m[k][0], gx, fmaf(m[k][1], gy, fmaf(m[k][2], gz, m[k][9])));
        float ny = fmaf(m[k][3], gx, fmaf(m[k][4], gy, fmaf(m[k][5], gz, m[k][10])));
        float nz = fmaf(m[k][6], gx, fmaf(m[k][7], gy, fmaf(m[k][8], gz, m[k][11])));
        gx = nx; gy = ny; gz = nz;
    }
    int ix = (int)((gx + 50.0f) / 0.5f);   // truncation toward zero == jnp astype(int32)
    int iy = (int)((gy + 50.0f) / 0.5f);
    int iz = (int)((gz + 10.0f) / 20.0f);  // IEEE divide: 1/20 not exactly representable
    if (ix >= 0 && ix < XD && iy >= 0 && iy < YD && iz == 0)
        return ix * YD + iy;
    return -1;
}

// ---------------------------------------------------------------------------
// Kernel 2: scatter-add into per-timestep accumulators tmp[t][voxel][ch].
// One wave per point: lane l owns channels {2l, 2l+1}. x load = one coalesced
// 256B nontemporal burst; skipped entirely for out-of-range points (~24%).
// ---------------------------------------------------------------------------
#define PW 8  // points per wave

template<int T>
__global__ void MOSEG_scatter_tmp(const float* __restrict__ x,
                                  const float* __restrict__ geom,
                                  const float* __restrict__ xf,
                                  float* __restrict__ tmp) {
    constexpr int NST = (S_T - 1) - T;
    const int lane = threadIdx.x & 31;
    const int wave = blockIdx.x * (blockDim.x >> 5) + (threadIdx.x >> 5);

    float m[NST > 0 ? NST : 1][12];
    for (int k = 0; k < NST; ++k)
        #pragma unroll
        for (int j = 0; j < 12; ++j) m[k][j] = xf[(T + k) * 12 + j];

    const float* xt  = x    + (size_t)T * NPTS * CCH;
    const float* gt  = geom + (size_t)T * NPTS * 3;
    float*       dst = tmp  + (size_t)T * CV;

    size_t p0 = (size_t)wave * PW;
    if (p0 >= (size_t)NPTS) return;
    __builtin_prefetch(gt + (p0 + 8 * PW) * 3, 0, 0);  // global_prefetch_b8

    #pragma unroll 1
    for (int i = 0; i < PW; ++i) {
        size_t p = p0 + i;
        float gx = gt[p*3 + 0], gy = gt[p*3 + 1], gz = gt[p*3 + 2];
        int v = voxel_index<NST>(m, gx, gy, gz);
        if (v >= 0) {
            f32x2 f = __builtin_nontemporal_load((const f32x2*)(xt + p * CCH + 2 * lane));
            float* d = dst + (size_t)v * CCH + 2 * lane;
            atomAddF(d + 0, f.x);
            atomAddF(d + 1, f.y);
        }
    }
}

// Fallback: signed atomics straight into d_out (channel-major), if ws is small.
template<int T>
__global__ void MOSEG_scatter_direct(const float* __restrict__ x,
                                     const float* __restrict__ geom,
                                     const float* __restrict__ xf,
                                     float* __restrict__ out) {
    constexpr int NST = (S_T - 1) - T;
    const int lane = threadIdx.x & 31;
    const int wave = blockIdx.x * (blockDim.x >> 5) + (threadIdx.x >> 5);

    float m[NST > 0 ? NST : 1][12];
    for (int k = 0; k < NST; ++k)
        #pragma unroll
        for (int j = 0; j < 12; ++j) m[k][j] = xf[(T + k) * 12 + j];

    const float* xt = x    + (size_t)T * NPTS * CCH;
    const float* gt = geom + (size_t)T * NPTS * 3;

    size_t p0 = (size_t)wave * PW;
    if (p0 >= (size_t)NPTS) return;

    #pragma unroll 1
    for (int i = 0; i < PW; ++i) {
        size_t p = p0 + i;
        float gx = gt[p*3 + 0], gy = gt[p*3 + 1], gz = gt[p*3 + 2];
        int v = voxel_index<NST>(m, gx, gy, gz);
        if (v >= 0) {
            f32x2 f = __builtin_nontemporal_load((const f32x2*)(xt + p * CCH + 2 * lane));
            float sgn = 2.0f;
            #pragma unroll
            for (int tt = T; tt < S_T; ++tt) {        // out_t += 2*(-1)^(t-T) * feat
                float* d = out + (size_t)tt * CV + (size_t)(2 * lane) * NVOX + v;
                atomAddF(d,        sgn * f.x);
                atomAddF(d + NVOX, sgn * f.y);
                sgn = -sgn;
            }
        }
    }
}

// ---------------------------------------------------------------------------
// Kernel 3: recurrence combine + (voxel,ch) -> (ch,voxel) transpose.
// CDNA5 path: stage the three 64x64 f32 tiles into LDS with
// GLOBAL_LOAD_ASYNC_TO_LDS_B128 (per-lane independent LDS destination lets the
// DMA write a padded-row layout directly), track with ASYNCcnt, then read
// transposed out of LDS and stream results with nontemporal stores.
// bev0=2a, bev1=2(b-a), bev2=2(c-(b-a))  [x2 is exact, matches reference FP]
// ---------------------------------------------------------------------------
#define RS 68  // padded row stride (floats): multiple of 4 keeps 16B alignment

__global__ void MOSEG_combine(const float* __restrict__ tmp, float* __restrict__ out) {
    __shared__ float smem[3 * 64 * RS];
    const int tid   = threadIdx.x;
    const int vbase = blockIdx.x * 64;

    const unsigned sbase = (unsigned)(size_t)(void*)smem;  // LDS byte offset (low 32 bits)

    #pragma unroll
    for (int pl = 0; pl < 3; ++pl) {
        const float* src = tmp + (size_t)pl * CV;
        unsigned tbase = sbase + (unsigned)(pl * 64 * RS * 4);
        #pragma unroll
        for (int it = 0; it < 4; ++it) {
            int e   = it * 256 + tid;          // 16-byte chunk index within tile
            int vl  = e >> 4;                  // local voxel row
            int ch4 = (e & 15) * 4;            // starting channel of the 4-float chunk
            const float* g = src + (((size_t)(vbase + vl)) << 6) + ch4;
            unsigned     l = tbase + (unsigned)(vl * RS + ch4) * 4u;
            asm volatile("global_load_async_to_lds_b128 %0, %1, off"
                         :: "v"(l), "v"(g) : "memory");
        }
    }
    asm volatile("s_wait_asynccnt 0" ::: "memory");
    __syncthreads();

    const float* s0 = smem;
    const float* s1 = smem + 64 * RS;
    const float* s2 = smem + 2 * 64 * RS;

    #pragma unroll
    for (int it = 0; it < 16; ++it) {
        int idx = it * 256 + tid;
        int ch = idx >> 6, vl = idx & 63;      // consecutive lanes -> consecutive v
        int so = vl * RS + ch;
        float a = s0[so], b = s1[so], c = s2[so];
        float bma = b - a;
        size_t o = (size_t)ch * NVOX + vbase + vl;   // coalesced output writes
        __builtin_nontemporal_store(a + a,            out + o);
        __builtin_nontemporal_store(bma * 2.0f,       out + CV + o);
        __builtin_nontemporal_store((c - bma) * 2.0f, out + 2 * CV + o);
    }
}

// ---------------------------------------------------------------------------
extern "C" void kernel_launch(void* const* d_in, const int* in_sizes, int n_in,
                              void* d_out, int out_size, void* d_ws, size_t ws_size,
                              hipStream_t stream) {
    (void)in_sizes; (void)n_in;
    const float* x    = (const float*)d_in[0];
    const float* geom = (const float*)d_in[1];
    const float* ego  = (const float*)d_in[2];
    float* out = (float*)d_out;
    float* xf  = (float*)d_ws;

    const size_t XF_BYTES  = 4096;
    const size_t TMP_BYTES = 3 * CV * sizeof(float);   // 30.72 MB, L2-resident

    MOSEG_xform<<<1, 32, 0, stream>>>(ego, xf);

    const int blocks = NPTS / (8 * PW);  // 8 waves/block * PW points/wave = 7560

    if (ws_size >= XF_BYTES + TMP_BYTES) {
        float* tmp = (float*)((char*)d_ws + XF_BYTES);
        hipMemsetAsync(tmp, 0, TMP_BYTES, stream);
        MOSEG_scatter_tmp<0><<<blocks, 256, 0, stream>>>(x, geom, xf, tmp);
        MOSEG_scatter_tmp<1><<<blocks, 256, 0, stream>>>(x, geom, xf, tmp);
        MOSEG_scatter_tmp<2><<<blocks, 256, 0, stream>>>(x, geom, xf, tmp);
        MOSEG_combine<<<NVOX / 64, 256, 0, stream>>>(tmp, out);
    } else {
        hipMemsetAsync(out, 0, (size_t)out_size * sizeof(float), stream);
        MOSEG_scatter_direct<0><<<blocks, 256, 0, stream>>>(x, geom, xf, out);
        MOSEG_scatter_direct<1><<<blocks, 256, 0, stream>>>(x, geom, xf, out);
        MOSEG_scatter_direct<2><<<blocks, 256, 0, stream>>>(x, geom, xf, out);
    }
}